// SingleInputPeptidePocketConvLayer_11072425689949
// MI455X (gfx1250) — compile-verified
//
#include <hip/hip_runtime.h>

typedef __attribute__((ext_vector_type(2))) float v2f;
typedef __attribute__((ext_vector_type(8))) float v8f;

namespace {
constexpr int kM = 15;            // peptide positions
constexpr int kR = 64;            // representation size
constexpr int kP = 34;            // pocket positions
constexpr int kF = 9;             // filter taps
constexpr int kOut = kR + kF - 1; // 72
constexpr int kChunks = kP * (kOut / 4); // 34*18 = 612 float4 chunks per sample

// LDS layout per sample (strides multiple-of-4 floats for b128 alignment,
// non-zero mod-64 row shift to spread banks)
constexpr int kAStride = 17;                   // 16 K cols padded (scalar access only)
constexpr int kARows   = 48;                   // 34 rows padded to 3 row-tiles
constexpr int kASize   = kARows * kAStride;    // 816 floats
constexpr int kBStride = 68;                   // 64 cols padded (mult of 4)
constexpr int kBRows   = 16;                   // 15 K rows padded
constexpr int kBSize   = kBRows * kBStride;    // 1088 floats
constexpr int kSStride = 84;                   // 8 zero | 64 data | 8 zero | 4 pad
constexpr int kSSize   = kP * kSStride;        // 2856 floats
constexpr int kFSize   = 308;                  // 34*9 taps padded to mult of 4
constexpr int kSampleLds = kASize + kBSize + kSSize + kFSize; // 5068 floats
constexpr int kWavesPerBlock = 2;              // 40,544 B static LDS / block
} // namespace

__global__ __launch_bounds__(kWavesPerBlock * 32) void
peptide_pocket_conv_wmma(const float* __restrict__ pep,   // [B,15,64]
                         const int*   __restrict__ aaidx, // [B,34]
                         const int*   __restrict__ adj,   // [B,34,15]
                         const float* __restrict__ kern,  // [20,9]
                         float* __restrict__ out,         // [B,34,72]
                         int nB) {
  __shared__ __align__(16) float lds[kWavesPerBlock * kSampleLds];

  const int lane = threadIdx.x & 31;
  const int wave = threadIdx.x >> 5;
  const int b = blockIdx.x * kWavesPerBlock + wave;
  if (b >= nB) return;

  float* Al = lds + wave * kSampleLds;       // offset 0      (mult of 4)
  float* Bl = Al + kASize;                   // offset 816
  float* Sl = Bl + kBSize;                   // offset 1904
  float* Fl = Sl + kSSize;                   // offset 4760
  const float4 z4 = make_float4(0.f, 0.f, 0.f, 0.f);

  // ---- Zero A (48x16 @17) with b128 stores: 816/4 = 204 float4 ----
  {
    float4* a4 = (float4*)Al;
    for (int t = lane; t < kASize / 4; t += 32) a4[t] = z4;
  }
  // ---- Stage A = adj_mask[b] as f32 (510 ints via exactly 255 b64 loads) ----
  {
    const int2* ab2 = (const int2*)(adj + (size_t)b * (kP * kM));
    for (int t2 = lane; t2 < (kP * kM) / 2; t2 += 32) {
      const int2 v = ab2[t2];
      const int g0 = 2 * t2, g1 = g0 + 1;
      const int p0 = g0 / kM, p1 = g1 / kM;
      Al[p0 * kAStride + (g0 - p0 * kM)] = (float)v.x;
      Al[p1 * kAStride + (g1 - p1 * kM)] = (float)v.y;
    }
  }

  // ---- Stage B = peptide_encoding[b] via b128: 240 float4 ----
  {
    const float4* pb4 = (const float4*)(pep + (size_t)b * (kM * kR));
    float4* b4 = (float4*)Bl;
    for (int t4 = lane; t4 < (kM * kR) / 4; t4 += 32) {
      const int m = t4 >> 4;       // row (16 float4 per 64-col row)
      const int c = t4 & 15;       // float4 col
      b4[m * (kBStride / 4) + c] = pb4[t4];
    }
    // zero K-pad row 15
    for (int c = lane; c < kR / 4; c += 32) b4[15 * (kBStride / 4) + c] = z4;
  }

  // ---- Zero the conv halo of S: per row float4 cols {0,1,18,19} ----
  {
    float4* s4 = (float4*)Sl;
    for (int t = lane; t < kP * 4; t += 32) {
      const int p = t >> 2;
      const int q = t & 3;
      s4[p * (kSStride / 4) + (q < 2 ? q : 16 + q)] = z4;
    }
  }

  // ---- Stage per-row filter taps: Fl[p*9+k] = kern[aaidx[b,p]*9+k] ----
  {
    const int* ib = aaidx + (size_t)b * kP;
    for (int t = lane; t < kP * kF; t += 32) {
      const int p = t / kF;
      const int k = t - p * kF;
      Fl[t] = kern[(size_t)ib[p] * kF + k];
    }
  }

  // ---- s = A(34x15) x B(15x64) via V_WMMA_F32_16X16X4_F32 ----
  const int laneLo = lane & 15;
  const int laneHi = lane >> 4;
  for (int rt = 0; rt < 3; ++rt) {
    const int arow = rt * 16 + laneLo;
    for (int ct = 0; ct < 4; ++ct) {
      const int ncol = ct * 16 + laneLo;
      v8f acc = {};
#pragma unroll
      for (int ks = 0; ks < 4; ++ks) {
        // 32-bit 16x4 A layout: lanes 0-15 hold K=k,k+1; lanes 16-31 K=k+2,k+3
        const int ka = ks * 4 + laneHi * 2;
        v2f av, bv;
        av.x = Al[arow * kAStride + ka];
        av.y = Al[arow * kAStride + ka + 1];
        bv.x = Bl[(ka + 0) * kBStride + ncol];
        bv.y = Bl[(ka + 1) * kBStride + ncol];
        acc = __builtin_amdgcn_wmma_f32_16x16x4_f32(
            false, av, false, bv, (short)0, acc, false, false);
      }
      // C layout: VGPR i -> M = rt*16 + laneHi*8 + i, N = ct*16 + laneLo
#pragma unroll
      for (int i = 0; i < 8; ++i) {
        const int prow = rt * 16 + laneHi * 8 + i;
        if (prow < kP) Sl[prow * kSStride + 8 + ncol] = acc[i];
      }
    }
  }

  // ---- Full conv: each lane emits one float4 output chunk per iter ----
  // chunk t = p*18 + c covers out[b, p, 4c .. 4c+3]; note t is exactly the
  // global float4 index within this sample's output.
  {
    const float4* s4 = (const float4*)Sl;
    float4* o4 = (float4*)(out + (size_t)b * kP * kOut);
    for (int t = lane; t < kChunks; t += 32) {
      const int p = t / 18;
      const int c = t - p * 18;
      // 12-float window: srow floats [4c-8 .. 4c+3] == float4 idx p*21 + c .. +2
      const int sbase = p * (kSStride / 4) + c;
      const float4 a0 = s4[sbase];
      const float4 a1 = s4[sbase + 1];
      const float4 a2 = s4[sbase + 2];
      float xv[12] = {a0.x, a0.y, a0.z, a0.w, a1.x, a1.y,
                      a1.z, a1.w, a2.x, a2.y, a2.z, a2.w};
      const float* fp = Fl + p * kF;
      float f[kF];
#pragma unroll
      for (int k = 0; k < kF; ++k) f[k] = fp[k];
      float o[4];
#pragma unroll
      for (int e = 0; e < 4; ++e) {
        float acc = 0.0f;
#pragma unroll
        for (int k = 0; k < kF; ++k) acc = fmaf(f[k], xv[8 + e - k], acc);
        o[e] = acc;
      }
      o4[t] = make_float4(o[0], o[1], o[2], o[3]); // global_store_b128
    }
  }
}

extern "C" void kernel_launch(void* const* d_in, const int* in_sizes, int n_in,
                              void* d_out, int out_size, void* d_ws, size_t ws_size,
                              hipStream_t stream) {
  const float* pep   = (const float*)d_in[0]; // peptide_encoding [B,15,64]
  const int*   aaidx = (const int*)d_in[1];   // pocket_aa_idx    [B,34]
  const int*   adj   = (const int*)d_in[2];   // adj_mask         [B,34,15]
  const float* kern  = (const float*)d_in[3]; // kernels          [20,9]
  float* out = (float*)d_out;                 // [B,34,72]

  const int nB = in_sizes[0] / (kM * kR);
  const int blocks = (nB + kWavesPerBlock - 1) / kWavesPerBlock;
  peptide_pocket_conv_wmma<<<blocks, kWavesPerBlock * 32, 0, stream>>>(
      pep, aaidx, adj, kern, out, nB);
}